// MultiHeadedAttention_88201448390825
// MI455X (gfx1250) — compile-verified
//
#include <hip/hip_runtime.h>

// CDNA5 (gfx1250) fused multi-head attention, fp32 WMMA path.
// D = A(16x4) * B(4x16) + C(16x16), all f32, wave32.
typedef float v2f __attribute__((ext_vector_type(2)));
typedef float v8f __attribute__((ext_vector_type(8)));

#define WMMA_F32(a, b, c) \
  __builtin_amdgcn_wmma_f32_16x16x4_f32(false, (a), false, (b), (short)0, (c), false, false)

constexpr int B_ = 8, S_ = 2048, D_ = 64, H_ = 4, HD_ = 16;
constexpr int ST_ = S_ / 16;  // 128 sixteen-row seq tiles

// Layout helpers (per ISA 7.12.2, 32-bit operands, wave32):
//  A 16x4 : lane n -> M = n%16, K = v + 2*(n/16)          (v = VGPR index 0..1)
//  B 4x16 : lane n -> K = v + 2*(n/16), N = n%16
//  C 16x16: lane n, vgpr r -> M = r + 8*(n/16), N = n%16

// ---------------------------------------------------------------------------
// Phase 1: Q/K/V projections. One wave per (b, seq-tile, head).
//   q_ws : [B,H,S,HD] row-major, pre-scaled by 1/sqrt(HD)
//   kT_ws: [B,H,HD,S]  (transposed: QK^T B-operand loads become contiguous)
//   v_ws : [B,H,S,HD] row-major
// ---------------------------------------------------------------------------
__global__ __launch_bounds__(32) void qkv_proj_kernel(
    const float* __restrict__ data, const float* __restrict__ Wq,
    const float* __restrict__ Wk, const float* __restrict__ Wv,
    float* __restrict__ q_ws, float* __restrict__ kT_ws, float* __restrict__ v_ws) {
  const int bid = blockIdx.x;
  const int h  = bid % H_;
  const int st = (bid / H_) % ST_;
  const int b  = bid / (H_ * ST_);
  const int lane = threadIdx.x, m = lane & 15, half = lane >> 4;
  const int s0 = st * 16;

  // A operand: data tile rows s0..s0+15, D=64 split into 16 K-chunks of 4.
  const float* dp = data + ((size_t)b * S_ + s0 + m) * D_;
  v2f a[16];
#pragma unroll
  for (int kc = 0; kc < 16; ++kc) {
    const int k0 = kc * 4 + 2 * half;
    a[kc][0] = dp[k0];
    a[kc][1] = dp[k0 + 1];
  }

  const float* wq = Wq + (size_t)h * D_ * HD_;
  const float* wk = Wk + (size_t)h * D_ * HD_;
  const float* wv = Wv + (size_t)h * D_ * HD_;

  v8f cq = {}, ck = {}, cv = {};
#pragma unroll
  for (int kc = 0; kc < 16; ++kc) {
    const int kr = kc * 4 + 2 * half;
    v2f bq, bk, bv;
    bq[0] = wq[kr * HD_ + m];       bq[1] = wq[(kr + 1) * HD_ + m];
    bk[0] = wk[kr * HD_ + m];       bk[1] = wk[(kr + 1) * HD_ + m];
    bv[0] = wv[kr * HD_ + m];       bv[1] = wv[(kr + 1) * HD_ + m];
    cq = WMMA_F32(a[kc], bq, cq);
    ck = WMMA_F32(a[kc], bk, ck);
    cv = WMMA_F32(a[kc], bv, cv);
  }

  const int bh = b * H_ + h;
  // Q: row-major, fold in softmax scale 1/sqrt(16) = 0.25.
  float* qp = q_ws + ((size_t)bh * S_ + s0 + 8 * half) * HD_ + m;
#pragma unroll
  for (int r = 0; r < 8; ++r) qp[r * HD_] = cq[r] * 0.25f;
  // K^T: element (hd=N=m, s=s0+M); consecutive r -> contiguous s: 128-bit stores.
  float* kp = kT_ws + ((size_t)bh * HD_ + m) * S_ + s0 + 8 * half;
  *reinterpret_cast<float4*>(kp)     = make_float4(ck[0], ck[1], ck[2], ck[3]);
  *reinterpret_cast<float4*>(kp + 4) = make_float4(ck[4], ck[5], ck[6], ck[7]);
  // V: row-major.
  float* vp = v_ws + ((size_t)bh * S_ + s0 + 8 * half) * HD_ + m;
#pragma unroll
  for (int r = 0; r < 8; ++r) vp[r * HD_] = cv[r];
}

// ---------------------------------------------------------------------------
// Phase 2: flash attention. One wave per (b, h, 16-row Q tile); stream 128 KV
// tiles; never materialize the 2048x2048 score matrix (saves ~1 GB of HBM
// round-trips vs naive). Online softmax state kept per C-layout row.
// ---------------------------------------------------------------------------
__global__ __launch_bounds__(32) void flash_attn_kernel(
    const float* __restrict__ q_ws, const float* __restrict__ kT_ws,
    const float* __restrict__ v_ws, float* __restrict__ cat_ws) {
  __shared__ float lds[16 * 17];  // padded 16x16 bounce buffer (C->A relayout)

  const int bid = blockIdx.x;
  const int qt = bid % ST_;
  const int bh = bid / ST_;
  const int b = bh / H_, h = bh % H_;
  const int lane = threadIdx.x, m = lane & 15, half = lane >> 4;
  const int s0 = qt * 16;

  // Q tile as A operand (HD=16 -> 4 K-chunks), already scaled.
  const float* qp = q_ws + ((size_t)bh * S_ + s0 + m) * HD_;
  v2f aq[4];
#pragma unroll
  for (int kc = 0; kc < 4; ++kc) {
    const int k0 = kc * 4 + 2 * half;
    aq[kc][0] = qp[k0];
    aq[kc][1] = qp[k0 + 1];
  }

  v8f acc = {};
  float mrow[8], lrow[8];
#pragma unroll
  for (int r = 0; r < 8; ++r) { mrow[r] = -3.0e38f; lrow[r] = 0.0f; }

  const float* ktb = kT_ws + (size_t)bh * HD_ * S_;
  const float* vb  = v_ws  + (size_t)bh * S_ * HD_;

  for (int kt = 0; kt < ST_; ++kt) {
    const int t0 = kt * 16;

    // scores = Qs * K^T  (16x16, f32 accum)
    v8f cs = {};
#pragma unroll
    for (int kc = 0; kc < 4; ++kc) {
      const int kr = kc * 4 + 2 * half;
      v2f bk;
      bk[0] = ktb[(size_t)kr * S_ + t0 + m];
      bk[1] = ktb[(size_t)(kr + 1) * S_ + t0 + m];
      cs = WMMA_F32(aq[kc], bk, cs);
    }

    // Online softmax: rows of C live across a 16-lane half -> butterfly
    // reductions with xor masks 1,2,4,8 stay inside the half.
    float pr[8];
#pragma unroll
    for (int r = 0; r < 8; ++r) {
      float sv = cs[r];
      float rmax = sv;
      rmax = fmaxf(rmax, __shfl_xor(rmax, 1, 32));
      rmax = fmaxf(rmax, __shfl_xor(rmax, 2, 32));
      rmax = fmaxf(rmax, __shfl_xor(rmax, 4, 32));
      rmax = fmaxf(rmax, __shfl_xor(rmax, 8, 32));
      const float mnew = fmaxf(mrow[r], rmax);
      const float corr = __expf(mrow[r] - mnew);
      const float p = __expf(sv - mnew);
      float rsum = p;
      rsum += __shfl_xor(rsum, 1, 32);
      rsum += __shfl_xor(rsum, 2, 32);
      rsum += __shfl_xor(rsum, 4, 32);
      rsum += __shfl_xor(rsum, 8, 32);
      lrow[r] = lrow[r] * corr + rsum;
      mrow[r] = mnew;
      acc[r] = acc[r] * corr;
      pr[r] = p;
    }

    // P: C-layout -> A-layout via LDS (single wave; DS ops are in-order).
#pragma unroll
    for (int r = 0; r < 8; ++r) lds[(r + 8 * half) * 17 + m] = pr[r];
    __syncthreads();
    v2f ap[4];
#pragma unroll
    for (int kc = 0; kc < 4; ++kc) {
      const int k0 = kc * 4 + 2 * half;
      ap[kc][0] = lds[m * 17 + k0];
      ap[kc][1] = lds[m * 17 + k0 + 1];
    }
    __syncthreads();

    // acc += P * V
#pragma unroll
    for (int kc = 0; kc < 4; ++kc) {
      const int kr = kc * 4 + 2 * half;
      v2f bv;
      bv[0] = vb[(size_t)(t0 + kr) * HD_ + m];
      bv[1] = vb[(size_t)(t0 + kr + 1) * HD_ + m];
      acc = WMMA_F32(ap[kc], bv, acc);
    }
  }

  // Normalize and scatter into concatenated-heads buffer [B,S,H*HD].
  float* op = cat_ws + ((size_t)b * S_ + s0 + 8 * half) * (H_ * HD_) + h * HD_ + m;
#pragma unroll
  for (int r = 0; r < 8; ++r) op[r * (H_ * HD_)] = acc[r] / lrow[r];
}

// ---------------------------------------------------------------------------
// Phase 3: output projection  out = cat[16384x64] @ Wo^T[64x64] + bo.
// One wave per 16-row tile; 4 column tiles x 16 K-chunks = 64 WMMAs.
// ---------------------------------------------------------------------------
__global__ __launch_bounds__(32) void out_proj_kernel(
    const float* __restrict__ cat_ws, const float* __restrict__ Wo,
    const float* __restrict__ bo, float* __restrict__ out) {
  const int i0 = blockIdx.x * 16;
  const int lane = threadIdx.x, m = lane & 15, half = lane >> 4;

  const float* cp = cat_ws + (size_t)(i0 + m) * D_;
  v2f a[16];
#pragma unroll
  for (int kc = 0; kc < 16; ++kc) {
    const int k0 = kc * 4 + 2 * half;
    a[kc][0] = cp[k0];
    a[kc][1] = cp[k0 + 1];
  }

#pragma unroll
  for (int jt = 0; jt < 4; ++jt) {
    const int j0 = jt * 16;
    v8f c = {};
    const float* wrow = Wo + (size_t)(j0 + m) * D_;  // B[k][j] = Wo[j][k]
#pragma unroll
    for (int kc = 0; kc < 16; ++kc) {
      const int kr = kc * 4 + 2 * half;
      v2f bw;
      bw[0] = wrow[kr];
      bw[1] = wrow[kr + 1];
      c = WMMA_F32(a[kc], bw, c);
    }
    const float bias = bo[j0 + m];
    float* op = out + (size_t)(i0 + 8 * half) * D_ + j0 + m;
#pragma unroll
    for (int r = 0; r < 8; ++r) op[r * D_] = c[r] + bias;
  }
}

// ---------------------------------------------------------------------------
extern "C" void kernel_launch(void* const* d_in, const int* in_sizes, int n_in,
                              void* d_out, int out_size, void* d_ws, size_t ws_size,
                              hipStream_t stream) {
  const float* data = (const float*)d_in[0];
  const float* Wq   = (const float*)d_in[1];
  const float* Wk   = (const float*)d_in[2];
  const float* Wv   = (const float*)d_in[3];
  const float* Wo   = (const float*)d_in[4];
  const float* bo   = (const float*)d_in[5];
  float* out = (float*)d_out;

  float* ws = (float*)d_ws;
  const size_t per = (size_t)B_ * H_ * S_ * HD_;  // 1,048,576 floats each
  float* q_ws   = ws;
  float* kT_ws  = ws + per;
  float* v_ws   = ws + 2 * per;
  float* cat_ws = ws + 3 * per;  // [B,S,H*HD]; total ws use: 16 MB

  qkv_proj_kernel<<<dim3(B_ * ST_ * H_), dim3(32), 0, stream>>>(
      data, Wq, Wk, Wv, q_ws, kT_ws, v_ws);
  flash_attn_kernel<<<dim3(B_ * H_ * ST_), dim3(32), 0, stream>>>(
      q_ws, kT_ws, v_ws, cat_ws);
  out_proj_kernel<<<dim3((B_ * S_) / 16), dim3(32), 0, stream>>>(
      cat_ws, Wo, bo, out);
}